// YOLOLoss_23089744183399
// MI455X (gfx1250) — compile-verified
//
#include <hip/hip_runtime.h>

// ---------------------------------------------------------------------------
// YOLO-v1 loss, MI455X (gfx1250, wave32).
// Streaming-bound: ~193MB read once -> ~8.3us floor at 23.3 TB/s.
// One cell-pair (240B, 16B-aligned) per thread -> all b128 nontemporal loads.
// Wave reduction via V_WMMA_F32_16X16X4_F32 (B = ones), deterministic
// two-pass block reduction through d_ws (no atomics).
// ---------------------------------------------------------------------------

typedef float f32x4 __attribute__((ext_vector_type(4)));
typedef float v2f   __attribute__((ext_vector_type(2)));
typedef float v8f   __attribute__((ext_vector_type(8)));

#define YOLO_S            56
#define YOLO_D            30
#define LAMBDA_COORD_F    5.0f
#define LAMBDA_NOOBJ_F    0.5f

// Sum of the 32 per-lane values, returned in every lane.
// A: a0 = partial, a1 = 0  ->  A[m][0]=partial[m], A[m][2]=partial[m+16]
// B: all ones (4x16)       ->  D[m][n] = partial[m] + partial[m+16], all n
// Lane n (0..15) holds D[0..7][n] in d0..d7; lane n+16 holds D[8..15][n].
// Per-lane sum of d0..d7 + shfl_xor(16) = full 32-lane total.
__device__ __forceinline__ float wave_sum_wmma(float partial) {
    v2f a; a[0] = partial; a[1] = 0.0f;
    v2f b; b[0] = 1.0f;    b[1] = 1.0f;
    v8f c = {};
    c = __builtin_amdgcn_wmma_f32_16x16x4_f32(
            /*neg_a=*/false, a, /*neg_b=*/false, b,
            /*c_mod=*/(short)0, c, /*reuse_a=*/false, /*reuse_b=*/false);
    float s = c[0] + c[1] + c[2] + c[3] + c[4] + c[5] + c[6] + c[7];
    s += __shfl_xor(s, 16, 32);
    return s;
}

__device__ __forceinline__ float cell_loss(const float* p, const float* t) {
    float d0 = p[0] - t[0];
    float d1 = p[1] - t[1];
    float dxy = d0 * d0 + d1 * d1;
    float sw = __builtin_sqrtf(p[2]) - __builtin_sqrtf(t[2]);
    float sh = __builtin_sqrtf(p[3]) - __builtin_sqrtf(t[3]);
    float dwh = sw * sw + sh * sh;
    float dc = p[4] - t[4];
    float conf = dc * dc;
    float cls = 0.0f;
#pragma unroll
    for (int i = 5; i < YOLO_D; ++i) {
        float d = p[i] - t[i];
        cls += d * d;
    }
    float obj = (t[4] == 1.0f) ? 1.0f : 0.0f;
    return obj * (LAMBDA_COORD_F * (dxy + dwh) + conf + cls)
         + (1.0f - obj) * (LAMBDA_NOOBJ_F * conf);
}

__global__ __launch_bounds__(256)
void yolo_cell_kernel(const float* __restrict__ pred,
                      const float* __restrict__ targ,
                      float* __restrict__ blocksums,
                      int ncells) {
    __shared__ float lds[8];
    const int npairs = ncells >> 1;
    const int pair = blockIdx.x * 256 + threadIdx.x;

    float per = 0.0f;
    if (pair < npairs) {
        // 240 bytes per pair -> 16B aligned -> 15 x b128 per tensor.
        const f32x4* __restrict__ p4 =
            (const f32x4*)(pred + (size_t)pair * (2 * YOLO_D));
        const f32x4* __restrict__ t4 =
            (const f32x4*)(targ + (size_t)pair * (2 * YOLO_D));
        float pb[2 * YOLO_D], tb[2 * YOLO_D];
#pragma unroll
        for (int i = 0; i < 15; ++i) {
            f32x4 pv = __builtin_nontemporal_load(p4 + i);
            f32x4 tv = __builtin_nontemporal_load(t4 + i);
#pragma unroll
            for (int j = 0; j < 4; ++j) {
                pb[4 * i + j] = pv[j];
                tb[4 * i + j] = tv[j];
            }
        }
        per = cell_loss(pb, tb) + cell_loss(pb + YOLO_D, tb + YOLO_D);
    }

    // Odd-cell tail (not hit for the reference shapes; kept for generality).
    if ((ncells & 1) && blockIdx.x == 0 && threadIdx.x == 0) {
        const float* p = pred + (size_t)(ncells - 1) * YOLO_D;
        const float* t = targ + (size_t)(ncells - 1) * YOLO_D;
        float pb[YOLO_D], tb[YOLO_D];
#pragma unroll
        for (int i = 0; i < YOLO_D; ++i) { pb[i] = p[i]; tb[i] = t[i]; }
        per += cell_loss(pb, tb);
    }

    // No lane has exited: EXEC is all ones here (WMMA requirement).
    float wsum = wave_sum_wmma(per);
    const int wid = threadIdx.x >> 5;
    if ((threadIdx.x & 31) == 0) lds[wid] = wsum;
    __syncthreads();
    if (threadIdx.x == 0) {
        float s = 0.0f;
#pragma unroll
        for (int w = 0; w < 8; ++w) s += lds[w];
        blocksums[blockIdx.x] = s;
    }
}

__global__ __launch_bounds__(256)
void yolo_final_kernel(const float* __restrict__ blocksums,
                       float* __restrict__ out,
                       int nblocks, float inv_batch) {
    __shared__ float lds[8];
    float partial = 0.0f;
    for (int i = threadIdx.x; i < nblocks; i += 256)
        partial += blocksums[i];

    float wsum = wave_sum_wmma(partial);
    const int wid = threadIdx.x >> 5;
    if ((threadIdx.x & 31) == 0) lds[wid] = wsum;
    __syncthreads();
    if (threadIdx.x == 0) {
        float s = 0.0f;
#pragma unroll
        for (int w = 0; w < 8; ++w) s += lds[w];
        out[0] = s * inv_batch;
    }
}

extern "C" void kernel_launch(void* const* d_in, const int* in_sizes, int n_in,
                              void* d_out, int out_size, void* d_ws, size_t ws_size,
                              hipStream_t stream) {
    (void)n_in; (void)out_size; (void)ws_size;
    const float* pred = (const float*)d_in[0];
    const float* targ = (const float*)d_in[1];
    float* out = (float*)d_out;
    float* ws  = (float*)d_ws;

    const long long total = in_sizes[0];            // BATCH * S * S * D
    const int ncells = (int)(total / YOLO_D);       // BATCH * S * S
    const int batch  = ncells / (YOLO_S * YOLO_S);  // BATCH
    const float inv_batch = 1.0f / (float)batch;

    const int npairs  = ncells >> 1;
    const int nblocks = (npairs + 255) / 256;       // 1568 for reference shape

    yolo_cell_kernel<<<nblocks, 256, 0, stream>>>(pred, targ, ws, ncells);
    yolo_final_kernel<<<1, 256, 0, stream>>>(ws, out, nblocks, inv_batch);
}